// AvgFIStateProbabilitiesPaulied_22024592293964
// MI455X (gfx1250) — compile-verified
//
#include <hip/hip_runtime.h>
#include <math.h>

// ---------------------------------------------------------------------------
// AvgFIStateProbabilitiesPaulied — analytic-derivative reformulation
//   embed_kernel  : pre-embed Paulis into WMMA B-operand layout (f32, 64KB)
//   eig_kernel    : per-b 8x8 complex Hermitian Jacobi eigh (f64) + amp/P/Phi
//   fisher_kernel : TDM-stages embedded Paulis into LDS, then per (b,q)
//                   M = V^H Pq V via embedded-real 16x16x16 WMMA
//                   (V_WMMA_F32_16X16X4_F32 x4 per matmul), f64 epilogue
//   reduce_kernel : I = [drives^2 ; 1]^T @ s  via WMMA f32 16x16x4 over K=512
// ---------------------------------------------------------------------------

typedef float v2f __attribute__((ext_vector_type(2)));
typedef float v8f __attribute__((ext_vector_type(8)));
typedef unsigned int v4u __attribute__((ext_vector_type(4)));
typedef int v4i __attribute__((ext_vector_type(4)));
typedef int v8i __attribute__((ext_vector_type(8)));

#define NDRIVES 4
#define DIM     8
#define LENP    64
#define BATCH   512

// per-b workspace layout (doubles): V[128] amp[16] invP[8] lam[8] Phi[128]
#define WSB        288
#define OFF_V      0
#define OFF_AMP    128
#define OFF_INVP   144
#define OFF_LAM    152
#define OFF_PHI    160

#define EMBQ_ELEMS (LENP * 4 * 32 * 2)   // 16384 floats = 64KB

// ---------------------------------------------------------------------------
// Kernel 0: embed paulies into per-lane WMMA B-operand layout:
//   embQ[q][kc][lane][jj] = embed(Pq)[4*kc + jj + 2*(lane>>4)][lane&15]
// where embed(C) = [[Cr, -Ci],[Ci, Cr]] (16x16 real)
// ---------------------------------------------------------------------------
__global__ void __launch_bounds__(256)
embed_kernel(const double* __restrict__ paulies, float* __restrict__ embQ)
{
    const int idx = blockIdx.x * 256 + threadIdx.x;   // 0..16383
    const int jj   = idx & 1;
    const int lane = (idx >> 1) & 31;
    const int kc   = (idx >> 6) & 3;
    const int q    = idx >> 8;
    const int half = lane >> 4;
    const int col  = lane & 15;
    const int r    = 4 * kc + jj + 2 * half;          // embedded row
    const int rr = r & 7, cc = col & 7;
    const double re = paulies[((size_t)q * 64 + rr * 8 + cc) * 2 + 0];
    const double im = paulies[((size_t)q * 64 + rr * 8 + cc) * 2 + 1];
    const bool top = r < 8, left = col < 8;
    float v;
    if (top == left) v = (float)re;
    else             v = top ? (float)(-im) : (float)im;
    embQ[idx] = v;
}

// ---------------------------------------------------------------------------
// Kernel 1: w = d@K + b ; H = sum_q w_q P_q ; Jacobi eigh ; amp/P/Phi outputs
// one block (64 threads) per batch element; matrix entry (i,j) per thread
// ---------------------------------------------------------------------------
__global__ void __launch_bounds__(64)
eig_kernel(const double* __restrict__ drives, const double* __restrict__ kern,
           const double* __restrict__ bias,   const double* __restrict__ paulies,
           double* __restrict__ ws)
{
    __shared__ double wsh[LENP];
    __shared__ double Hr[8][8], Hi[8][8];
    __shared__ double Vr[8][8], Vi[8][8];
    __shared__ double rot[3];

    const int b = blockIdx.x;
    const int t = threadIdx.x;
    const int i = t >> 3, j = t & 7;

    // w[t] = sum_k drives[b,k]*kernel[k,t] + bias[t]
    double acc = bias[t];
    for (int k = 0; k < NDRIVES; ++k)
        acc += drives[(size_t)b * NDRIVES + k] * kern[(size_t)k * LENP + t];
    wsh[t] = acc;
    __syncthreads();

    // H[i][j] = sum_q w[q] * pauli[q][i][j]
    double hr = 0.0, hi = 0.0;
    for (int q = 0; q < LENP; ++q) {
        const double* p = paulies + ((size_t)q * 64 + t) * 2;
        hr += wsh[q] * p[0];
        hi += wsh[q] * p[1];
    }
    Hr[i][j] = hr; Hi[i][j] = hi;
    Vr[i][j] = (i == j) ? 1.0 : 0.0;
    Vi[i][j] = 0.0;
    __syncthreads();

    // cyclic complex Jacobi sweeps
    for (int sweep = 0; sweep < 8; ++sweep) {
        for (int p = 0; p < 8; ++p) {
            for (int q = p + 1; q < 8; ++q) {
                if (t == 0) {
                    double app = Hr[p][p], aqq = Hr[q][q];
                    double hre = Hr[p][q], him = Hi[p][q];
                    double ah = sqrt(hre * hre + him * him);
                    double c = 1.0, sr = 0.0, si = 0.0;
                    if (ah > 1e-300) {
                        double phr = hre / ah, phi = him / ah;
                        double tau = (aqq - app) / (2.0 * ah);
                        double tt  = (tau >= 0.0 ? 1.0 : -1.0) /
                                     (fabs(tau) + sqrt(1.0 + tau * tau));
                        c = 1.0 / sqrt(1.0 + tt * tt);
                        double srot = tt * c;
                        sr = srot * phr; si = srot * phi;
                    }
                    rot[0] = c; rot[1] = sr; rot[2] = si;
                }
                __syncthreads();
                const double c = rot[0], sr = rot[1], si = rot[2];
                // H <- H G  (column update)
                double vpr = Hr[i][p], vpi = Hi[i][p];
                double vqr = Hr[i][q], vqi = Hi[i][q];
                __syncthreads();
                if (j == p) { Hr[i][p] = c*vpr + sr*vqr - si*vqi;
                              Hi[i][p] = c*vpi + sr*vqi + si*vqr; }
                if (j == q) { Hr[i][q] = c*vqr - (sr*vpr + si*vpi);
                              Hi[i][q] = c*vqi - (sr*vpi - si*vpr); }
                __syncthreads();
                // H <- G^H H  (row update)
                double upr = Hr[p][j], upi = Hi[p][j];
                double uqr = Hr[q][j], uqi = Hi[q][j];
                __syncthreads();
                if (i == p) { Hr[p][j] = c*upr + sr*uqr + si*uqi;
                              Hi[p][j] = c*upi + sr*uqi - si*uqr; }
                if (i == q) { Hr[q][j] = c*uqr - (sr*upr - si*upi);
                              Hi[q][j] = c*uqi - (sr*upi + si*upr); }
                __syncthreads();
                // V <- V G
                double wpr = Vr[i][p], wpi = Vi[i][p];
                double wqr = Vr[i][q], wqi = Vi[i][q];
                __syncthreads();
                if (j == p) { Vr[i][p] = c*wpr + sr*wqr - si*wqi;
                              Vi[i][p] = c*wpi + sr*wqi + si*wqr; }
                if (j == q) { Vr[i][q] = c*wqr - (sr*wpr + si*wpi);
                              Vi[i][q] = c*wqi - (sr*wpi - si*wpr); }
                __syncthreads();
            }
        }
    }

    double* wb = ws + (size_t)b * WSB;
    wb[OFF_V + (i * 8 + j) * 2 + 0] = Vr[i][j];
    wb[OFF_V + (i * 8 + j) * 2 + 1] = Vi[i][j];
    if (t < 8) wb[OFF_LAM + t] = Hr[t][t];

    if (t < 8) {
        // amp_j = sum_s V[j][s] * e^{-i lam_s} * conj(V[0][s])
        double ar = 0.0, ai = 0.0;
        for (int s = 0; s < 8; ++s) {
            double lam = Hr[s][s];
            double er = cos(lam), ei = -sin(lam);
            double ur = Vr[0][s], ui = -Vi[0][s];
            double cr = er * ur - ei * ui;
            double ci = er * ui + ei * ur;
            ar += Vr[t][s] * cr - Vi[t][s] * ci;
            ai += Vr[t][s] * ci + Vi[t][s] * cr;
        }
        wb[OFF_AMP + t * 2 + 0] = ar;
        wb[OFF_AMP + t * 2 + 1] = ai;
        double P = ar * ar + ai * ai;
        wb[OFF_INVP + t] = 1.0 / P;
    }

    // Loewner matrix Phi_ij for the Frechet derivative of exp(-iH)
    {
        double li = Hr[i][i], lj = Hr[j][j];
        double eir = cos(li), eii = -sin(li);
        double ejr = cos(lj), eji = -sin(lj);
        double dl = li - lj;
        double pr, pi;
        if (fabs(dl) > 1e-9) { pr = (eir - ejr) / dl; pi = (eii - eji) / dl; }
        else                 { pr = eii;              pi = -eir; }  // -i e^{-i li}
        wb[OFF_PHI + (i * 8 + j) * 2 + 0] = pr;
        wb[OFF_PHI + (i * 8 + j) * 2 + 1] = pi;
    }
}

// embedded-real fetch for V:  V~ = [[Vr, -Vi],[Vi, Vr]]
__device__ __forceinline__ float fetchVemb(const float (*Vf)[8][2], int r, int c, bool hermT)
{
    const int rr = r & 7, cc = c & 7;
    float re, im;
    if (hermT) { re = Vf[cc][rr][0]; im = -Vf[cc][rr][1]; }   // conj transpose
    else       { re = Vf[rr][cc][0]; im =  Vf[rr][cc][1]; }
    const bool top = r < 8, left = c < 8;
    if (top == left) return re;
    return top ? -im : im;
}

// ---------------------------------------------------------------------------
// Kernel 2: per (b,q): M = V^H Pq V as embedded real 16x16x16 (WMMA f32 k=4 x4
// per matmul, x2 matmuls); f64 epilogue -> s[b][q] = sum_j dP_j^2 / P_j
// block = 1 batch element, 8 waves, each wave owns 8 Pauli directions.
// Embedded Pauli operands staged global->LDS via the Tensor Data Mover.
// ---------------------------------------------------------------------------
__global__ void __launch_bounds__(256)
fisher_kernel(const float* __restrict__ embQ, const double* __restrict__ ws,
              float* __restrict__ sbuf)
{
    const int b    = blockIdx.x;
    const int tid  = threadIdx.x;
    const int wave = tid >> 5;
    const int lane = tid & 31;
    const int half = lane >> 4;
    const int col  = lane & 15;

    const double* wb = ws + (size_t)b * WSB;

    __shared__ float  ldsQ[EMBQ_ELEMS];     // 64KB: TDM destination
    __shared__ float  Vf[8][8][2];          // V as f32 (re,im) for WMMA operands
    __shared__ double Vd[8][8][2];          // V in f64 for the epilogue
    __shared__ double PhiS[8][8][2];
    __shared__ double ampS[8][2];
    __shared__ double invPS[8];
    __shared__ float  M1buf[8][16][16];     // per-wave 16x16 embedded M1
    __shared__ float  Mbuf[8][8][8][2];     // per-wave complex M
    __shared__ double tbuf[8][8][2];
    __shared__ double valbuf[8][8];

    // ---- TDM: DMA the 64KB embedded-Pauli block into LDS (wave 0 issues) ----
    if (wave == 0) {
        const unsigned long long ga = (unsigned long long)(uintptr_t)embQ;
        const unsigned int lds_base = (unsigned int)(uintptr_t)&ldsQ[0];
        // D# group0: count=1, lds_addr, 57-bit global addr, type=2 ("image")
        v4u g0 = { 1u,
                   lds_base,
                   (unsigned int)(ga & 0xFFFFFFFFull),
                   (unsigned int)((ga >> 32) & 0x01FFFFFFull) | 0x80000000u };
        // D# group1: wg_mask=0, data_size=4B(code 2), dims: 16384 x 1 tile/tensor
        v8i g1 = { (int)(2u << 16),            // data_size=2 (4 bytes)
                   (int)(16384u << 16),        // tensor_dim0 = 16384 (lo16 @63:48)
                   (int)(1u << 16),            // tensor_dim1 = 1     (@95:80)
                   (int)(16384u << 16),        // tile_dim0 = 16384   (@127:112)
                   1,                          // tile_dim1 = 1       (@143:128)
                   16384,                      // tensor_dim0_stride  (@191:160)
                   0, 0 };
        v4i gz4 = { 0, 0, 0, 0 };              // groups 2/3 unused (2D tensor)
        v8i gz8 = { 0, 0, 0, 0, 0, 0, 0, 0 };
        __builtin_amdgcn_tensor_load_to_lds(g0, g1, gz4, gz4, gz8, 0);
        __builtin_amdgcn_s_wait_tensorcnt((short)0);
    }

    if (tid < 64) {
        const int i = tid >> 3, j = tid & 7;
        const double vr = wb[OFF_V + (i * 8 + j) * 2 + 0];
        const double vi = wb[OFF_V + (i * 8 + j) * 2 + 1];
        Vf[i][j][0] = (float)vr;  Vf[i][j][1] = (float)vi;
        Vd[i][j][0] = vr;         Vd[i][j][1] = vi;
        PhiS[i][j][0] = wb[OFF_PHI + (i * 8 + j) * 2 + 0];
        PhiS[i][j][1] = wb[OFF_PHI + (i * 8 + j) * 2 + 1];
    }
    if (tid < 8) {
        ampS[tid][0] = wb[OFF_AMP + tid * 2 + 0];
        ampS[tid][1] = wb[OFF_AMP + tid * 2 + 1];
        invPS[tid]   = wb[OFF_INVP + tid];
    }
    __syncthreads();

    // hoist the V operands: A = embed(V^H) in A-layout, B = embed(V) in B-layout
    v2f aVH[4], bV[4];
    for (int kc = 0; kc < 4; ++kc)
        for (int jj = 0; jj < 2; ++jj) {
            const int kIdx = 4 * kc + jj + 2 * half;
            aVH[kc][jj] = fetchVemb(Vf, col, kIdx, true);
            bV[kc][jj]  = fetchVemb(Vf, kIdx, col, false);
        }

    for (int it = 0; it < 8; ++it) {
        const int q = wave * 8 + it;

        // matmul1: M1~ = embed(V^H) * embed(Pq); B operands from LDS (ds_load_b64)
        v8f d1 = {0.f, 0.f, 0.f, 0.f, 0.f, 0.f, 0.f, 0.f};
        for (int kc = 0; kc < 4; ++kc) {
            const v2f bq = *(const v2f*)&ldsQ[((q * 4 + kc) * 32 + lane) * 2];
            d1 = __builtin_amdgcn_wmma_f32_16x16x4_f32(
                     false, aVH[kc], false, bq, (short)0, d1, false, false);
        }

        __syncthreads();                         // protect vs previous-iter reads
        for (int v = 0; v < 8; ++v)
            M1buf[wave][v + 8 * half][col] = d1[v];
        __syncthreads();

        // matmul2: M~ = M1~ * embed(V)
        v8f d2 = {0.f, 0.f, 0.f, 0.f, 0.f, 0.f, 0.f, 0.f};
        for (int kc = 0; kc < 4; ++kc) {
            v2f am;
            for (int jj = 0; jj < 2; ++jj)
                am[jj] = M1buf[wave][col][4 * kc + jj + 2 * half];
            d2 = __builtin_amdgcn_wmma_f32_16x16x4_f32(
                     false, am, false, bV[kc], (short)0, d2, false, false);
        }

        // extract complex M: rows 0-7 / cols 0-7 = Re, rows 8-15 / cols 0-7 = Im
        if (lane < 8)
            for (int v = 0; v < 8; ++v) Mbuf[wave][v][lane][0] = d2[v];
        else if (lane >= 16 && lane < 24)
            for (int v = 0; v < 8; ++v) Mbuf[wave][v][lane - 16][1] = d2[v];
        __syncthreads();

        // t_r = sum_c Phi[r][c] * M[r][c] * conj(V[0][c])
        if (lane < 8) {
            const int r = lane;
            double tr = 0.0, ti = 0.0;
            for (int c = 0; c < 8; ++c) {
                const double mr = (double)Mbuf[wave][r][c][0];
                const double mi = (double)Mbuf[wave][r][c][1];
                const double fr = PhiS[r][c][0], fi = PhiS[r][c][1];
                const double gr = fr * mr - fi * mi;
                const double gi = fr * mi + fi * mr;
                const double ur = Vd[0][c][0], ui = -Vd[0][c][1];
                tr += gr * ur - gi * ui;
                ti += gr * ui + gi * ur;
            }
            tbuf[wave][r][0] = tr; tbuf[wave][r][1] = ti;
        }
        __syncthreads();

        // damp_j = sum_s V[j][s] t_s ; dP_j = 2 Re(conj(amp_j) damp_j)
        if (lane < 8) {
            const int j = lane;
            double dr = 0.0, di = 0.0;
            for (int s = 0; s < 8; ++s) {
                const double vr = Vd[j][s][0], vi = Vd[j][s][1];
                const double tr = tbuf[wave][s][0], ti = tbuf[wave][s][1];
                dr += vr * tr - vi * ti;
                di += vr * ti + vi * tr;
            }
            const double dP = 2.0 * (ampS[j][0] * dr + ampS[j][1] * di);
            valbuf[wave][j] = dP * dP * invPS[j];
        }
        __syncthreads();

        if (lane == 0) {
            double sa = 0.0;
            for (int j = 0; j < 8; ++j) sa += valbuf[wave][j];
            sbuf[(size_t)b * LENP + q] = (float)sa;
        }
    }
}

// ---------------------------------------------------------------------------
// Kernel 3: I = [drives^2 (4 rows) ; ones (1 row) ; 0] @ s  via WMMA over K=512
// 1 block, 4 waves; wave w owns output columns [16w, 16w+16)
// out[0:256] = I_k (row-major [4][64]),  out[256:320] = I_b
// ---------------------------------------------------------------------------
__global__ void __launch_bounds__(128)
reduce_kernel(const double* __restrict__ drives, const float* __restrict__ sbuf,
              float* __restrict__ out)
{
    const int tid  = threadIdx.x;
    const int wave = tid >> 5;
    const int lane = tid & 31;
    const int half = lane >> 4;
    const int col  = lane & 15;
    const int ntile = wave * 16;

    // stage drives^2 (f32) into LDS so the K-loop is branch-free
    __shared__ float d2s[BATCH * NDRIVES];   // 8KB
    for (int e = tid; e < BATCH * NDRIVES; e += 128) {
        const double dd = drives[e];
        d2s[e] = (float)(dd * dd);
    }
    __syncthreads();

    v8f acc = {0.f, 0.f, 0.f, 0.f, 0.f, 0.f, 0.f, 0.f};
    for (int kc = 0; kc < BATCH / 4; ++kc) {
        v2f a, bb;
        for (int jj = 0; jj < 2; ++jj) {
            const int k = 4 * kc + jj + 2 * half;
            const int m = col;                       // A row
            a[jj]  = (m < NDRIVES) ? d2s[k * NDRIVES + m]
                                   : ((m == NDRIVES) ? 1.0f : 0.0f);
            bb[jj] = sbuf[(size_t)k * LENP + ntile + col];
        }
        acc = __builtin_amdgcn_wmma_f32_16x16x4_f32(
                  false, a, false, bb, (short)0, acc, false, false);
    }

    if (half == 0) {
        const int n = ntile + col;
        for (int v = 0; v < NDRIVES + 1; ++v) {      // rows 0..4 live in vgprs 0..4
            const float val = acc[v] * (1.0f / (float)BATCH);
            if (v < NDRIVES) out[v * LENP + n] = val;
            else             out[NDRIVES * LENP + n] = val;
        }
    }
}

// ---------------------------------------------------------------------------
extern "C" void kernel_launch(void* const* d_in, const int* in_sizes, int n_in,
                              void* d_out, int out_size, void* d_ws, size_t ws_size,
                              hipStream_t stream)
{
    (void)in_sizes; (void)n_in; (void)out_size; (void)ws_size;
    const double* x       = (const double*)d_in[0]; (void)x;   // unused by math
    const double* drives  = (const double*)d_in[1];            // [512,4]
    const double* kern    = (const double*)d_in[2];            // [4,64]
    const double* bias    = (const double*)d_in[3];            // [64]
    const double* paulies = (const double*)d_in[4];            // [64,8,8] c128
    float* out = (float*)d_out;                                // [320]

    double* ws_eig = (double*)d_ws;                            // 512*288 doubles
    float*  sbuf   = (float*)((char*)d_ws + (size_t)BATCH * WSB * sizeof(double));
    float*  embQ   = sbuf + (size_t)BATCH * LENP;              // 16384 floats

    embed_kernel<<<EMBQ_ELEMS / 256, 256, 0, stream>>>(paulies, embQ);
    eig_kernel<<<BATCH, 64, 0, stream>>>(drives, kern, bias, paulies, ws_eig);
    fisher_kernel<<<BATCH, 256, 0, stream>>>(embQ, ws_eig, sbuf);
    reduce_kernel<<<1, 128, 0, stream>>>(drives, sbuf, out);
}